// GraphConvModule_34772055229173
// MI455X (gfx1250) — compile-verified
//
#include <hip/hip_runtime.h>

typedef __attribute__((ext_vector_type(2))) float v2f;
typedef __attribute__((ext_vector_type(8))) float v8f;

#define D 128
#define LSTR 132        // LDS row stride (floats): 132 mod 64 = 4 -> fragment lanes spread banks
#define BPACK_FLOATS (D * D)   // packed weight: 8 col-tiles * 32 ksteps * 32 lanes * 2 floats

// ---------------- utility: zero a float buffer ----------------
__global__ __launch_bounds__(256) void zero_kernel(float* __restrict__ p, long n) {
    long i = (long)blockIdx.x * blockDim.x + threadIdx.x;
    long stride = (long)gridDim.x * blockDim.x;
    for (; i < n; i += stride) p[i] = 0.0f;
}

// ---------------- edge norm pipeline ----------------
__global__ __launch_bounds__(256) void deg_kernel(const long long* __restrict__ src,
                                                  const long long* __restrict__ dst,
                                                  float* __restrict__ deg, int E) {
    int e = blockIdx.x * blockDim.x + threadIdx.x;
    if (e >= E) return;
    long long s = src[e], d = dst[e];
    if (s != d) atomicAdd(deg + s, 1.0f);
}

__global__ __launch_bounds__(256) void dis_kernel(float* __restrict__ deg, int N) {
    int i = blockIdx.x * blockDim.x + threadIdx.x;
    if (i >= N) return;
    float dg = deg[i];
    deg[i] = (dg > 0.0f) ? rsqrtf(fmaxf(dg, 1.0f)) : 0.0f;
}

__global__ __launch_bounds__(256) void norm_kernel(const long long* __restrict__ src,
                                                   const long long* __restrict__ dst,
                                                   const float* __restrict__ dis,
                                                   float* __restrict__ norm, int E) {
    int e = blockIdx.x * blockDim.x + threadIdx.x;
    if (e >= E) return;
    long long s = src[e], d = dst[e];
    norm[e] = (s != d) ? dis[s] * dis[d] : 0.0f;
}

// ---------------- pack weight into per-lane WMMA B-fragment order ----------------
// Bp[((c*32 + kk)*32 + lane)] (v2f) = { W[kk*4 + 2*hi][c*16 + lr], W[kk*4 + 1 + 2*hi][c*16 + lr] }
// so the GEMM can fetch a whole wave's B fragment with one coalesced global_load_b64.
__global__ __launch_bounds__(256) void pack_b_kernel(const float* __restrict__ W,
                                                     float* __restrict__ Bp) {
    int i = blockIdx.x * blockDim.x + threadIdx.x;   // 0 .. 8191
    if (i >= 8 * 32 * 32) return;
    int lane = i & 31;
    int kk   = (i >> 5) & 31;
    int c    = i >> 10;
    int lr   = lane & 15;
    int hi   = lane >> 4;
    int k    = kk * 4 + 2 * hi;
    v2f b;
    b.x = W[k * D + c * 16 + lr];
    b.y = W[(k + 1) * D + c * 16 + lr];
    *(v2f*)(Bp + (size_t)i * 2) = b;
}

// ---------------- scatter: T[dst] += -norm_e * X[src] ----------------
// One wave (32 lanes) per edge; each lane owns 4 features (float4 gather, 4 f32 atomics).
// Destination rows are L2-resident (105 MB working set < 192 MB L2).
__global__ __launch_bounds__(256) void scatter_kernel(const float* __restrict__ X,
                                                      const long long* __restrict__ src,
                                                      const long long* __restrict__ dst,
                                                      const float* __restrict__ norm,
                                                      float* __restrict__ T, int E) {
    long gt = (long)blockIdx.x * blockDim.x + threadIdx.x;
    int e = (int)(gt >> 5);
    int lane = (int)(gt & 31);
    if (e >= E) return;
    float w = -norm[e];
    if (w == 0.0f) return;              // self-loop / isolated: contributes nothing
    int s = (int)src[e];
    int d = (int)dst[e];
    const float4 xv = *(const float4*)(X + (long)s * D + lane * 4);
    float* tp = T + (long)d * D + lane * 4;
    atomicAdd(tp + 0, w * xv.x);
    atomicAdd(tp + 1, w * xv.y);
    atomicAdd(tp + 2, w * xv.z);
    atomicAdd(tp + 3, w * xv.w);
}

// ---------------- fused dual-GEMM + bias + ReLU via WMMA f32 16x16x4 ----------------
// out = relu(A @ W0 + T @ W1 + bias). Block = 8 waves; block stages the shared 16x128
// A and T tiles into LDS once; wave wv owns the 16x16 output tile at col wv*16.
// B comes from the packed layout: one coalesced global_load_b64 per fragment.
// Inner loop: 2x ds_load_2addr_b64 + 2x global_load_b64 + 2x v_wmma (indep. accumulators).
__global__ __launch_bounds__(256) void cheb_gemm_kernel(const float* __restrict__ A,
                                                        const float* __restrict__ T,
                                                        const float* __restrict__ B0p,
                                                        const float* __restrict__ B1p,
                                                        const float* __restrict__ bias,
                                                        float* __restrict__ out) {
    __shared__ float lA[16 * LSTR];
    __shared__ float lT[16 * LSTR];

    const int t    = threadIdx.x;
    const int lane = t & 31;
    const int wv   = t >> 5;
    const int row0 = blockIdx.x * 16;
    const int col0 = wv * 16;
    const int lr   = lane & 15;   // A: row in tile; B/C/D: col in tile
    const int hi   = lane >> 4;   // lane group selects K-pair / M-half

    // ---- stage A/T tiles (16x128 each) into LDS, cooperatively ----
    {
        const float* Ab = A + (long)row0 * D;
        const float* Tb = T + (long)row0 * D;
#pragma unroll
        for (int i = t; i < 16 * 32; i += 256) {   // 512 float4 chunks, 2 per thread
            int r  = i >> 5;
            int cg = i & 31;
            float4 va = *(const float4*)(Ab + r * D + cg * 4);
            float4 vt = *(const float4*)(Tb + r * D + cg * 4);
            *(float4*)(&lA[r * LSTR + cg * 4]) = va;
            *(float4*)(&lT[r * LSTR + cg * 4]) = vt;
        }
    }
    __syncthreads();

    // ---- main loop: LDS A/T fragments + packed coalesced B, two accumulators ----
    v8f c0 = {};
    v8f c1 = {};
    const float* la  = &lA[lr * LSTR + 2 * hi];
    const float* lt  = &lT[lr * LSTR + 2 * hi];
    const v2f*   bp0 = (const v2f*)B0p + (size_t)wv * 32 * 32 + lane;
    const v2f*   bp1 = (const v2f*)B1p + (size_t)wv * 32 * 32 + lane;
#pragma unroll
    for (int kk = 0; kk < 32; ++kk) {
        v2f ax = *(const v2f*)(la + kk * 4);
        v2f at = *(const v2f*)(lt + kk * 4);
        v2f b0 = bp0[kk * 32];
        v2f b1 = bp1[kk * 32];
        c0 = __builtin_amdgcn_wmma_f32_16x16x4_f32(false, ax, false, b0, (short)0, c0, false, false);
        c1 = __builtin_amdgcn_wmma_f32_16x16x4_f32(false, at, false, b1, (short)0, c1, false, false);
    }

    // ---- epilogue: combine, bias, ReLU, store ----
    const float bv = bias[col0 + lr];
#pragma unroll
    for (int v = 0; v < 8; ++v) {
        int m = v + 8 * hi;                       // C/D layout: VGPR v -> row v (+8 for hi lanes)
        float r = c0[v] + c1[v] + bv;
        out[(long)(row0 + m) * D + col0 + lr] = (r > 0.0f) ? r : 0.0f;
    }
}

// ---------------- launch ----------------
extern "C" void kernel_launch(void* const* d_in, const int* in_sizes, int n_in,
                              void* d_out, int out_size, void* d_ws, size_t ws_size,
                              hipStream_t stream) {
    const float*     x    = (const float*)d_in[0];
    const long long* ei   = (const long long*)d_in[1];   // int64 (2, E)
    const float*     W1_0 = (const float*)d_in[2];
    const float*     W1_1 = (const float*)d_in[3];
    const float*     b1   = (const float*)d_in[4];
    const float*     W2_0 = (const float*)d_in[5];
    const float*     W2_1 = (const float*)d_in[6];
    const float*     b2   = (const float*)d_in[7];
    float* out = (float*)d_out;

    const int N = in_sizes[0] / D;       // 100000
    const int E = in_sizes[1] / 2;       // 640000
    const long long* src = ei;
    const long long* dst = ei + E;

    // workspace layout (floats):
    // deg/dis[N] | norm[E] | Tx[N*D] | h[N*D] | Bp1_0 | Bp1_1 | Bp2_0 | Bp2_1
    float* ws    = (float*)d_ws;
    float* deg   = ws;
    float* norm  = deg + N;
    float* Tx    = norm + E;
    float* h     = Tx + (size_t)N * D;
    float* Bp1_0 = h + (size_t)N * D;
    float* Bp1_1 = Bp1_0 + BPACK_FLOATS;
    float* Bp2_0 = Bp1_1 + BPACK_FLOATS;
    float* Bp2_1 = Bp2_0 + BPACK_FLOATS;

    const int TB = 256;
    const int eblocks   = (E + TB - 1) / TB;
    const int nblocks   = (N + TB - 1) / TB;
    const long scatterT = (long)E * 32;
    const int  sblocks  = (int)((scatterT + TB - 1) / TB);
    const long nd       = (long)N * D;
    const int  zblocks  = 4096;
    const int  gblocks  = N / 16;        // 6250, exact
    const int  pblocks  = (8 * 32 * 32 + TB - 1) / TB;

    // edge norms
    zero_kernel<<<nblocks, TB, 0, stream>>>(deg, (long)N);
    deg_kernel<<<eblocks, TB, 0, stream>>>(src, dst, deg, E);
    dis_kernel<<<nblocks, TB, 0, stream>>>(deg, N);
    norm_kernel<<<eblocks, TB, 0, stream>>>(src, dst, deg, norm, E);

    // pack the 4 weight matrices into WMMA B-fragment order (once per launch)
    pack_b_kernel<<<pblocks, TB, 0, stream>>>(W1_0, Bp1_0);
    pack_b_kernel<<<pblocks, TB, 0, stream>>>(W1_1, Bp1_1);
    pack_b_kernel<<<pblocks, TB, 0, stream>>>(W2_0, Bp2_0);
    pack_b_kernel<<<pblocks, TB, 0, stream>>>(W2_1, Bp2_1);

    // layer 1: Tx = -scatter(norm * x[src]); h = relu(x@W1_0 + Tx@W1_1 + b1)
    zero_kernel<<<zblocks, TB, 0, stream>>>(Tx, nd);
    scatter_kernel<<<sblocks, TB, 0, stream>>>(x, src, dst, norm, Tx, E);
    cheb_gemm_kernel<<<gblocks, TB, 0, stream>>>(x, Tx, Bp1_0, Bp1_1, b1, h);

    // layer 2: Tx = -scatter(norm * h[src]); out = relu(h@W2_0 + Tx@W2_1 + b2)
    zero_kernel<<<zblocks, TB, 0, stream>>>(Tx, nd);
    scatter_kernel<<<sblocks, TB, 0, stream>>>(h, src, dst, norm, Tx, E);
    cheb_gemm_kernel<<<gblocks, TB, 0, stream>>>(h, Tx, Bp2_0, Bp2_1, b2, out);
}